// SequenceWavKAN_58213986730190
// MI455X (gfx1250) — compile-verified
//
#include <hip/hip_runtime.h>
#include <hip/hip_bf16.h>

// ---------------------------------------------------------------------------
// SequenceWavKAN on MI455X (gfx1250): fp32 end-to-end, all GEMMs through
// V_WMMA_F32_16X16X4_F32 (wave32 fragments: A,B = 2 f32 VGPRs, C/D = 8).
// GRU recurrence keeps h, gh and w_hh resident in LDS and prefetches the
// next step's input-gate row (global_prefetch_b8) to shorten the serial
// dependence chain.
// ---------------------------------------------------------------------------

#define IN_F   360
#define HID    64
#define NCLS   5
#define BATCH  64
#define SEQ    128
#define NTOT   (BATCH * SEQ)          // 8192 samples
#define G3     (3 * HID)              // 192 gates
#define MH_C   0.8673250705840776f    // 2 / (sqrt(3) * pi^0.25)

typedef __attribute__((ext_vector_type(2))) float v2f;
typedef __attribute__((ext_vector_type(8))) float v8f;

static __device__ __forceinline__ v8f wmma_f32(v2f a, v2f b, v8f c) {
    // D = A(16x4) * B(4x16) + C(16x16), all f32
    return __builtin_amdgcn_wmma_f32_16x16x4_f32(
        /*neg_a=*/false, a, /*neg_b=*/false, b,
        /*c_mod=*/(short)0, c, /*reuse_a=*/false, /*reuse_b=*/false);
}

static __device__ __forceinline__ float sigm(float x) {
    return 1.0f / (1.0f + __expf(-x));
}

// ---------------------------------------------------------------------------
// Kernel 1: WavKAN (mexican hat) + base(silu) GEMV + BatchNorm + LayerNorm.
// One block per sample n (= b*SEQ + s, x_flat order), 64 threads = 64 outputs.
// ---------------------------------------------------------------------------
__global__ __launch_bounds__(64) void wavkan_kernel(
    const float* __restrict__ x,
    const float* __restrict__ wav_scale, const float* __restrict__ wav_trans,
    const float* __restrict__ wav_w,     const float* __restrict__ base_w,
    const float* __restrict__ bng, const float* __restrict__ bnb,
    const float* __restrict__ bnm, const float* __restrict__ bnv,
    const float* __restrict__ lng, const float* __restrict__ lnb,
    float* __restrict__ xemb)
{
    __shared__ float xs[IN_F];
    __shared__ float ss[IN_F];   // silu(x)
    __shared__ float red[HID];

    const int n   = blockIdx.x;
    const int tid = threadIdx.x;
    const float* xr = x + (size_t)n * IN_F;

    for (int i = tid; i < IN_F; i += 64) {
        float v = xr[i];
        xs[i] = v;
        ss[i] = v * sigm(v);
    }
    __syncthreads();

    const int o = tid;
    const float* scr = wav_scale + (size_t)o * IN_F;
    const float* trr = wav_trans + (size_t)o * IN_F;
    const float* wwr = wav_w     + (size_t)o * IN_F;
    const float* bwr = base_w    + (size_t)o * IN_F;

    float accw = 0.0f, accb = 0.0f;
    #pragma unroll 4
    for (int i = 0; i < IN_F; ++i) {
        float t  = (xs[i] - trr[i]) / scr[i];
        float t2 = t * t;
        accw = fmaf((t2 - 1.0f) * __expf(-0.5f * t2), wwr[i], accw);
        accb = fmaf(ss[i], bwr[i], accb);
    }
    float comb = MH_C * accw + accb;
    // BatchNorm (eval)
    float xb = (comb - bnm[o]) * rsqrtf(bnv[o] + 1e-5f) * bng[o] + bnb[o];

    // LayerNorm over the 64 outputs
    red[o] = xb;
    __syncthreads();
    for (int off = 32; off > 0; off >>= 1) {
        if (tid < off) red[tid] += red[tid + off];
        __syncthreads();
    }
    float mu = red[0] * (1.0f / 64.0f);
    __syncthreads();
    float dv = xb - mu;
    red[o] = dv * dv;
    __syncthreads();
    for (int off = 32; off > 0; off >>= 1) {
        if (tid < off) red[tid] += red[tid + off];
        __syncthreads();
    }
    float var = red[0] * (1.0f / 64.0f);
    xemb[(size_t)n * HID + o] = dv * rsqrtf(var + 1e-5f) * lng[o] + lnb[o];
}

// ---------------------------------------------------------------------------
// Kernel 2: GRU input gates, both directions.
// gi[d][s][b][g] = x_sbf[s,b,:] . w_ih[g,:] + b_ih[g]
// WMMA GEMM: M = S*B = 8192 (m = s*64 + b), N = 192, K = 64.
// One wave (32 threads) per 16x16 tile. grid = (512, 12, 2).
// ---------------------------------------------------------------------------
__global__ __launch_bounds__(32) void gi_kernel(
    const float* __restrict__ xemb,
    const float* __restrict__ wih_f, const float* __restrict__ bih_f,
    const float* __restrict__ wih_b, const float* __restrict__ bih_b,
    float* __restrict__ gi)
{
    const int m0  = blockIdx.x * 16;
    const int n0  = blockIdx.y * 16;
    const int dir = blockIdx.z;
    const float* wih = dir ? wih_b : wih_f;
    const float* bih = dir ? bih_b : bih_f;
    float* gout = gi + (size_t)dir * NTOT * G3;

    const int lane = threadIdx.x;
    const int l    = lane & 15;
    const bool hi  = lane >= 16;

    const int m = m0 + l;             // m = s*64 + b
    const int s = m >> 6;
    const int b = m & 63;
    const float* arow = xemb + ((size_t)b * SEQ + s) * HID;   // x_flat row gather
    const int n = n0 + l;
    const float* brow = wih + (size_t)n * HID;                // B[k][n] = w_ih[n][k]

    v8f acc;
    float bias = bih[n];
    #pragma unroll
    for (int r = 0; r < 8; ++r) acc[r] = bias;

    #pragma unroll
    for (int k = 0; k < HID; k += 4) {
        int kk = k + (hi ? 2 : 0);
        v2f a, bb;
        a.x  = arow[kk];  a.y  = arow[kk + 1];
        bb.x = brow[kk];  bb.y = brow[kk + 1];
        acc = wmma_f32(a, bb, acc);
    }

    #pragma unroll
    for (int r = 0; r < 8; ++r) {
        int mg = m0 + r + (hi ? 8 : 0);
        gout[(size_t)mg * G3 + n] = acc[r];
    }
}

// ---------------------------------------------------------------------------
// Kernel 3: GRU recurrence. grid = 2 blocks (fwd/bwd), 256 threads = 8 waves.
// Per step: gh = h @ w_hh^T + b_hh via 48 WMMA tiles (6 per wave), then
// elementwise gate update. h, gh and w_hh live in LDS (16+48+48 = 112 KB,
// well under the 320 KB/WGP budget). The next step's gi row is prefetched
// (global_prefetch_b8) while the WMMA tiles execute.
// ---------------------------------------------------------------------------
__global__ __launch_bounds__(256) void gru_kernel(
    const float* __restrict__ gi,
    const float* __restrict__ whh_f, const float* __restrict__ bhh_f,
    const float* __restrict__ whh_b, const float* __restrict__ bhh_b,
    float* __restrict__ h_fwd, float* __restrict__ h_bwd)
{
    __shared__ float hlds[BATCH * HID];   // h[b][j]
    __shared__ float ghlds[BATCH * G3];   // gh[b][g]
    __shared__ float wlds[G3 * HID];      // w_hh[g][k] (B-matrix source)

    const int dir = blockIdx.x;
    const float* whh = dir ? whh_b : whh_f;
    const float* bhh = dir ? bhh_b : bhh_f;
    const float* gid = gi + (size_t)dir * NTOT * G3;
    float* hout = dir ? h_bwd : h_fwd;

    const int tid  = threadIdx.x;
    const int wid  = tid >> 5;
    const int lane = tid & 31;
    const int l    = lane & 15;
    const bool hi  = lane >= 16;

    for (int idx = tid; idx < G3 * HID; idx += 256) wlds[idx] = whh[idx];
    for (int idx = tid; idx < BATCH * HID; idx += 256) hlds[idx] = 0.0f;
    __syncthreads();

    for (int step = 0; step < SEQ; ++step) {
        const int s = dir ? (SEQ - 1 - step) : step;
        const int snext = dir ? (step + 1 < SEQ ? SEQ - 2 - step : s)
                              : (step + 1 < SEQ ? step + 1 : s);

        // Prefetch next step's gi row (48 KB) into cache while WMMAs run:
        // one cacheline-granular prefetch per 64 floats (256 B).
        {
            const float* gpre = gid + (size_t)snext * BATCH * G3;
            for (int off = tid * 64; off < BATCH * G3; off += 256 * 64)
                __builtin_prefetch(gpre + off, 0, 1);
        }

        // gh = h @ w_hh^T + b_hh  — 48 tiles (M=64 -> 4, N=192 -> 12)
        #pragma unroll
        for (int i = 0; i < 6; ++i) {
            int t  = wid * 6 + i;
            int m0 = (t & 3) * 16;
            int n0 = (t >> 2) * 16;
            int n  = n0 + l;
            const float* brow = wlds + n * HID;   // B[k][n] = w_hh[n][k]

            v8f acc;
            float bias = bhh[n];
            #pragma unroll
            for (int r = 0; r < 8; ++r) acc[r] = bias;

            #pragma unroll
            for (int k = 0; k < HID; k += 4) {
                int kk = k + (hi ? 2 : 0);
                v2f a, bb;
                a.x  = hlds[(m0 + l) * HID + kk];
                a.y  = hlds[(m0 + l) * HID + kk + 1];
                bb.x = brow[kk];
                bb.y = brow[kk + 1];
                acc = wmma_f32(a, bb, acc);
            }
            #pragma unroll
            for (int r = 0; r < 8; ++r)
                ghlds[(m0 + r + (hi ? 8 : 0)) * G3 + n] = acc[r];
        }
        __syncthreads();

        // gate update: 4096 (b,j) entries, 16 per thread
        const float* girow = gid + (size_t)s * BATCH * G3;
        float hnew[16];
        int q = 0;
        for (int idx = tid; idx < BATCH * HID; idx += 256, ++q) {
            int b = idx >> 6, j = idx & 63;
            const float* gib = girow + (size_t)b * G3;
            const float* ghb = ghlds + b * G3;
            float r  = sigm(gib[j]          + ghb[j]);
            float z  = sigm(gib[HID + j]    + ghb[HID + j]);
            float nn = tanhf(gib[2*HID + j] + r * ghb[2*HID + j]);
            hnew[q] = (1.0f - z) * nn + z * hlds[idx];
        }
        __syncthreads();
        q = 0;
        for (int idx = tid; idx < BATCH * HID; idx += 256, ++q) hlds[idx] = hnew[q];
        __syncthreads();
    }

    for (int idx = tid; idx < BATCH * HID; idx += 256) hout[idx] = hlds[idx];
}

// ---------------------------------------------------------------------------
// Kernel 4: classifier. fc1 (64x128 @ 128x64) via WMMA + bias + ReLU,
// fc2 (64x64 @ 64x5) by VALU. One block, 256 threads = 8 waves.
// ---------------------------------------------------------------------------
__global__ __launch_bounds__(256) void cls_kernel(
    const float* __restrict__ h_fwd, const float* __restrict__ h_bwd,
    const float* __restrict__ fc1w, const float* __restrict__ fc1b,
    const float* __restrict__ fc2w, const float* __restrict__ fc2b,
    float* __restrict__ out)
{
    __shared__ float hid[BATCH * 2 * HID];   // [b][0..127] = [h_fwd | h_bwd]
    __shared__ float act[BATCH * HID];       // relu(fc1)

    const int tid = threadIdx.x;
    for (int idx = tid; idx < BATCH * HID; idx += 256) {
        int b = idx >> 6, j = idx & 63;
        hid[b * 2 * HID + j]       = h_fwd[idx];
        hid[b * 2 * HID + HID + j] = h_bwd[idx];
    }
    __syncthreads();

    const int wid  = tid >> 5;
    const int lane = tid & 31;
    const int l    = lane & 15;
    const bool hi  = lane >= 16;

    // 16 tiles (M=64 -> 4, N=64 -> 4), 2 per wave, K = 128
    #pragma unroll
    for (int i = 0; i < 2; ++i) {
        int t  = wid * 2 + i;
        int m0 = (t & 3) * 16;
        int n0 = (t >> 2) * 16;
        int n  = n0 + l;
        const float* brow = fc1w + (size_t)n * (2 * HID);   // B[k][n] = fc1_w[n][k]

        v8f acc;
        float bias = fc1b[n];
        #pragma unroll
        for (int r = 0; r < 8; ++r) acc[r] = bias;

        #pragma unroll
        for (int k = 0; k < 2 * HID; k += 4) {
            int kk = k + (hi ? 2 : 0);
            v2f a, bb;
            a.x  = hid[(m0 + l) * (2 * HID) + kk];
            a.y  = hid[(m0 + l) * (2 * HID) + kk + 1];
            bb.x = brow[kk];
            bb.y = brow[kk + 1];
            acc = wmma_f32(a, bb, acc);
        }
        #pragma unroll
        for (int r = 0; r < 8; ++r) {
            int mg = m0 + r + (hi ? 8 : 0);
            float v = acc[r];
            act[mg * HID + n] = v > 0.0f ? v : 0.0f;
        }
    }
    __syncthreads();

    if (tid < BATCH * NCLS) {
        int b = tid / NCLS, c = tid % NCLS;
        float s = fc2b[c];
        #pragma unroll 8
        for (int j = 0; j < HID; ++j)
            s = fmaf(act[b * HID + j], fc2w[c * HID + j], s);
        out[b * NCLS + c] = s;
    }
}

// ---------------------------------------------------------------------------
extern "C" void kernel_launch(void* const* d_in, const int* in_sizes, int n_in,
                              void* d_out, int out_size, void* d_ws, size_t ws_size,
                              hipStream_t stream) {
    (void)in_sizes; (void)n_in; (void)out_size; (void)ws_size;

    const float* x_seq  = (const float*)d_in[0];
    const float* wsc    = (const float*)d_in[1];
    const float* wtr    = (const float*)d_in[2];
    const float* wwt    = (const float*)d_in[3];
    const float* bwt    = (const float*)d_in[4];
    const float* bng    = (const float*)d_in[5];
    const float* bnb    = (const float*)d_in[6];
    const float* bnm    = (const float*)d_in[7];
    const float* bnv    = (const float*)d_in[8];
    const float* lng    = (const float*)d_in[9];
    const float* lnb    = (const float*)d_in[10];
    const float* wih_f  = (const float*)d_in[11];
    const float* whh_f  = (const float*)d_in[12];
    const float* bih_f  = (const float*)d_in[13];
    const float* bhh_f  = (const float*)d_in[14];
    const float* wih_b  = (const float*)d_in[15];
    const float* whh_b  = (const float*)d_in[16];
    const float* bih_b  = (const float*)d_in[17];
    const float* bhh_b  = (const float*)d_in[18];
    const float* fc1w   = (const float*)d_in[19];
    const float* fc1b   = (const float*)d_in[20];
    const float* fc2w   = (const float*)d_in[21];
    const float* fc2b   = (const float*)d_in[22];

    float* ws   = (float*)d_ws;
    float* xemb = ws;                                   // 8192*64
    float* gi   = xemb + (size_t)NTOT * HID;            // 2*8192*192
    float* hf   = gi + (size_t)2 * NTOT * G3;           // 64*64
    float* hb   = hf + BATCH * HID;                     // 64*64

    wavkan_kernel<<<NTOT, 64, 0, stream>>>(
        x_seq, wsc, wtr, wwt, bwt, bng, bnb, bnm, bnv, lng, lnb, xemb);

    dim3 g2(NTOT / 16, G3 / 16, 2);
    gi_kernel<<<g2, 32, 0, stream>>>(xemb, wih_f, bih_f, wih_b, bih_b, gi);

    gru_kernel<<<2, 256, 0, stream>>>(gi, whh_f, bhh_f, whh_b, bhh_b, hf, hb);

    cls_kernel<<<1, 256, 0, stream>>>(hf, hb, fc1w, fc1b, fc2w, fc2b, (float*)d_out);
}